// MoeLayer_42855183680017
// MI455X (gfx1250) — compile-verified
//
#include <hip/hip_runtime.h>
#include <hip/hip_bf16.h>

typedef __attribute__((ext_vector_type(2))) float v2f;
typedef __attribute__((ext_vector_type(4))) float v4f;
typedef __attribute__((ext_vector_type(8))) float v8f;

#define NTOK 8192
#define NDIM 4096
#define NEXP 8

// ---------------------------------------------------------------------------
// Kernel 1: stream 134 MB of zeros into the results region (memory-bound).
// Nontemporal B128 stores, 128 bytes per thread, fully coalesced.
// ---------------------------------------------------------------------------
__global__ void __launch_bounds__(256) moe_zero_kernel(float* __restrict__ out) {
  const size_t base = (size_t)blockIdx.x * 2048u + threadIdx.x;  // in float4 units
  v4f* p = (v4f*)out;
  v4f z = {0.f, 0.f, 0.f, 0.f};
#pragma unroll
  for (int k = 0; k < 8; ++k) {
    __builtin_nontemporal_store(z, p + base + (size_t)k * 256u);
  }
}

// Select floats {2j, 2j+1} out of a 4x v4f (16-float) register chunk.
// j is always a compile-time constant after unrolling.
__device__ __forceinline__ v2f pick2(const v4f* c, int j) {
  v4f v = c[j >> 1];
  return (j & 1) ? (v2f){v.z, v.w} : (v2f){v.x, v.y};
}

// ---------------------------------------------------------------------------
// Kernel 2: gate logits via V_WMMA_F32_16X16X4_F32 + top-2 + softmax.
// One wave32 owns 32 tokens (two 16x16 C tiles sharing one B stream).
//
// K-slot -> dim mapping per macro-iteration (K=32 actual dims):
//   lane half h (h = lane>>4) supplies K-slots {2h, 2h+1} of every WMMA and
//   loads the contiguous 16-float chunk x[row][k0 + 16h .. k0 + 16h + 15]
//   as 4x global_load_b128. WMMA j (j=0..7) consumes floats {2j, 2j+1} of
//   each lane's chunk, for both A and B, so A/B agree on the K mapping.
//
// B (4x16): column n = lane&15; lanes with n>=8 redundantly load expert n&7
// (uniform, branch-free) -> C columns 8..15 are duplicates, never read.
// C (16x16 f32, 8 VGPRs): VGPR r -> M = r + 8*h, N = lane&15.
// ---------------------------------------------------------------------------
__global__ void __launch_bounds__(256) moe_gate_topk_kernel(
    const float* __restrict__ x,      // [NTOK, NDIM]
    const float* __restrict__ gw,     // [NEXP, NDIM]
    float* __restrict__ wout,         // [NTOK, 2] softmax weights
    float* __restrict__ iout)         // [NTOK, 2] expert indices (as float)
{
  __shared__ float logits[8][32][17];  // [wave][token][expert], padded

  const int lane = threadIdx.x & 31;
  const int wave = threadIdx.x >> 5;
  const int tokBase = (blockIdx.x * 8 + wave) * 32;   // 32 tokens per wave

  const int half = lane >> 4;
  const int l15  = lane & 15;

  const float* aptr0 = x  + (size_t)(tokBase + l15)      * NDIM + 16 * half;
  const float* aptr1 = x  + (size_t)(tokBase + 16 + l15) * NDIM + 16 * half;
  const float* bptr  = gw + (size_t)(l15 & 7)            * NDIM + 16 * half;

  v8f acc0 = {0.f, 0.f, 0.f, 0.f, 0.f, 0.f, 0.f, 0.f};
  v8f acc1 = {0.f, 0.f, 0.f, 0.f, 0.f, 0.f, 0.f, 0.f};

  for (int k0 = 0; k0 < NDIM; k0 += 32) {
    v4f a0[4], a1[4], b[4];
#pragma unroll
    for (int j = 0; j < 4; ++j) {
      a0[j] = *(const v4f*)(aptr0 + k0 + 4 * j);
      a1[j] = *(const v4f*)(aptr1 + k0 + 4 * j);
      b[j]  = *(const v4f*)(bptr  + k0 + 4 * j);
    }
#pragma unroll
    for (int j = 0; j < 8; ++j) {
      v2f bj = pick2(b, j);
      acc0 = __builtin_amdgcn_wmma_f32_16x16x4_f32(
          false, pick2(a0, j), false, bj, (short)0, acc0, false, false);
      acc1 = __builtin_amdgcn_wmma_f32_16x16x4_f32(
          false, pick2(a1, j), false, bj, (short)0, acc1, false, false);
    }
  }

  // Scatter both C tiles to LDS: acc[r] = logit(token r + 8*half, expert l15)
#pragma unroll
  for (int r = 0; r < 8; ++r) {
    logits[wave][r + 8 * half][l15]      = acc0[r];
    logits[wave][16 + r + 8 * half][l15] = acc1[r];
  }
  __syncthreads();

  // One lane per token (all 32 lanes active, branch-free):
  // top-2 with first-index tie-break (matches jax.lax.top_k) + fp32 softmax
  const float* lg = logits[wave][lane];
  float best = -3.402823466e+38f;
  int bi = 0;
#pragma unroll
  for (int e = 0; e < NEXP; ++e) {
    float v = lg[e];
    if (v > best) { best = v; bi = e; }
  }
  float best2 = -3.402823466e+38f;
  int bi2 = 0;
#pragma unroll
  for (int e = 0; e < NEXP; ++e) {
    if (e == bi) continue;
    float v = lg[e];
    if (v > best2) { best2 = v; bi2 = e; }
  }
  // softmax over [best, best2]: w0 = 1/(1+e^{d}), d = best2-best <= 0
  float d  = __expf(best2 - best);
  float w0 = 1.0f / (1.0f + d);
  float w1 = d * w0;

  const int t = tokBase + lane;
  *(v2f*)(wout + 2 * t) = (v2f){w0, w1};
  *(v2f*)(iout + 2 * t) = (v2f){(float)bi, (float)bi2};
}

// ---------------------------------------------------------------------------
// Launch: outputs concatenated flat: results[8192*4096] | weights[8192*2]
// | selected_experts[8192*2] (stored as float).
// ---------------------------------------------------------------------------
extern "C" void kernel_launch(void* const* d_in, const int* in_sizes, int n_in,
                              void* d_out, int out_size, void* d_ws, size_t ws_size,
                              hipStream_t stream) {
  (void)in_sizes; (void)n_in; (void)out_size; (void)d_ws; (void)ws_size;

  const float* x  = (const float*)d_in[0];   // [8192, 4096] fp32
  const float* gw = (const float*)d_in[1];   // [8, 4096] fp32

  float* out  = (float*)d_out;
  float* res  = out;                                   // [8192*4096] zeros
  float* wout = out + (size_t)NTOK * NDIM;             // [8192*2]
  float* iout = wout + (size_t)NTOK * 2;               // [8192*2]

  // Zero-fill results: 8,388,608 float4 / (256 thr * 8 per thr) = 4096 blocks
  moe_zero_kernel<<<4096, 256, 0, stream>>>(res);

  // Gate: 8192 tokens / (8 waves * 32 tokens) = 32 blocks of 256 threads
  moe_gate_topk_kernel<<<32, 256, 0, stream>>>(x, gw, wout, iout);
}